// BasicBlock_32375463477659
// MI455X (gfx1250) — compile-verified
//
#include <hip/hip_runtime.h>
#include <hip/hip_bf16.h>

typedef float v2f __attribute__((ext_vector_type(2)));
typedef float v8f __attribute__((ext_vector_type(8)));

#define D  128  // feature dim (both layers), known from reference
#define RT 4    // 16-row tiles per wave (register-blocked over B panel)

// ---------------------------------------------------------------------------
// Degree / normalization kernels
// ---------------------------------------------------------------------------
__global__ void k_init_deg(float* __restrict__ deg, int N) {
    int i = blockIdx.x * blockDim.x + threadIdx.x;
    if (i < N) deg[i] = 1.0f;  // self-loop contributes 1
}

__global__ void k_count_deg(const long long* __restrict__ ei, float* __restrict__ deg,
                            int E) {
    int e = blockIdx.x * blockDim.x + threadIdx.x;
    if (e < E) {
        int c = (int)ei[E + e];  // col = edge_index[1][e]
        atomicAdd(&deg[c], 1.0f);
    }
}

__global__ void k_rsqrt(float* __restrict__ deg, int N) {
    int i = blockIdx.x * blockDim.x + threadIdx.x;
    if (i < N) deg[i] = rsqrtf(deg[i]);  // deg >= 1 always
}

// ---------------------------------------------------------------------------
// fp32 WMMA GEMM: H[N x 128] = X[N x 128] * W[128 x 128]
// One wave32 per (64-row x 16-col) output block. The 128x16 B panel of W is
// preloaded into 64 VGPRs once, then 4 row tiles are swept with
// V_WMMA_F32_16X16X4_F32 (K stepped by 4) -> ~1 VMEM b64 load per WMMA.
//
// Out-of-range rows: A-row M only contributes to D-row M, so loads are merely
// clamped to a valid address (no zeroing needed) and the store loop guards
// rows >= N. This keeps the inner loop free of per-load cndmasks.
// ---------------------------------------------------------------------------
__global__ void k_gemm_wmma_f32(const float* __restrict__ X,
                                const float* __restrict__ W,
                                float* __restrict__ H, int N) {
    const int row0 = blockIdx.x * (16 * RT);
    const int n0   = blockIdx.y * 16;
    const int lane = threadIdx.x;        // 0..31
    const int half = lane >> 4;          // 0: K=k0,k0+1  1: K=k0+2,k0+3
    const int l15  = lane & 15;

    // --- preload B panel (W columns n0..n0+15, all K) into registers ---
    // B 4x16 f32 layout: VGPR0 = K k0 (lanes 0-15) / k0+2 (lanes 16-31),
    //                    VGPR1 = K k0+1 / k0+3.
    const float* wcol = W + n0 + l15 + (half ? 2 * D : 0);
    v2f breg[D / 4];
#pragma unroll
    for (int kk = 0; kk < D / 4; ++kk) {
        breg[kk].x = wcol[(size_t)(kk * 4)     * D];
        breg[kk].y = wcol[(size_t)(kk * 4 + 1) * D];
    }

    // --- sweep RT row tiles, reusing the B registers ---
#pragma unroll
    for (int t = 0; t < RT; ++t) {
        const int rb = row0 + t * 16;
        const int r  = rb + l15;
        const int rc = r < N ? r : (N - 1);  // clamp only; stores are guarded
        // A 16x4 f32 layout: lanes 0-15 read (K=k0,k0+1) of row r,
        //                    lanes 16-31 read (K=k0+2,k0+3) -> contiguous b64.
        const float* xrow = X + (size_t)rc * D + (half ? 2 : 0);

        v8f acc = {};
#pragma unroll
        for (int kk = 0; kk < D / 4; ++kk) {
            v2f a = *(const v2f*)(xrow + kk * 4);
            acc = __builtin_amdgcn_wmma_f32_16x16x4_f32(
                /*neg_a=*/false, a, /*neg_b=*/false, breg[kk],
                /*c_mod=*/(short)0, acc, /*reuse_a=*/false, /*reuse_b=*/false);
        }

        // D layout: VGPR v -> row rb+v (lanes 0-15), row rb+v+8 (lanes 16-31)
#pragma unroll
        for (int v = 0; v < 8; ++v) {
            int rr = rb + v + (half ? 8 : 0);
            if (rr < N) H[(size_t)rr * D + n0 + l15] = acc[v];
        }
    }
}

// ---------------------------------------------------------------------------
// y[i][j] = bias[j] + dis[i]^2 * h[i][j]   (self-loop message + bias)
// ---------------------------------------------------------------------------
__global__ void k_init_out(const float* __restrict__ h, const float* __restrict__ dis,
                           const float* __restrict__ bias, float* __restrict__ y,
                           int N) {
    int t = blockIdx.x * blockDim.x + threadIdx.x;
    if (t >= N * D) return;
    int i = t >> 7;
    int j = t & (D - 1);
    float d = dis[i];
    y[t] = bias[j] + d * d * h[t];
}

// ---------------------------------------------------------------------------
// Edge scatter: one wave per edge; each lane handles 4 consecutive features.
// out[col] += dis[row]*dis[col] * h[row]
// ---------------------------------------------------------------------------
__global__ void k_scatter(const float* __restrict__ h, const long long* __restrict__ ei,
                          const float* __restrict__ dis, float* __restrict__ out,
                          int E) {
    int t = blockIdx.x * blockDim.x + threadIdx.x;
    int e = t >> 5;
    int lane = t & 31;
    if (e >= E) return;
    int r = (int)ei[e];
    int c = (int)ei[E + e];
    float norm = dis[r] * dis[c];
    const float4 v = *(const float4*)(h + (size_t)r * D + lane * 4);
    float* o = out + (size_t)c * D + lane * 4;
    atomicAdd(o + 0, v.x * norm);
    atomicAdd(o + 1, v.y * norm);
    atomicAdd(o + 2, v.z * norm);
    atomicAdd(o + 3, v.w * norm);
}

__global__ void k_relu(float* __restrict__ y, int n) {
    int t = blockIdx.x * blockDim.x + threadIdx.x;
    if (t < n) y[t] = fmaxf(y[t], 0.0f);
}

// ---------------------------------------------------------------------------
// Launcher
// ---------------------------------------------------------------------------
extern "C" void kernel_launch(void* const* d_in, const int* in_sizes, int n_in,
                              void* d_out, int out_size, void* d_ws, size_t ws_size,
                              hipStream_t stream) {
    const float*     x   = (const float*)d_in[0];
    const long long* ei  = (const long long*)d_in[1];  // int64 [2, E]
    const float*     W1  = (const float*)d_in[2];
    const float*     b1  = (const float*)d_in[3];
    const float*     W2  = (const float*)d_in[4];
    const float*     b2  = (const float*)d_in[5];
    float*           out = (float*)d_out;

    const int N = in_sizes[0] / D;
    const int E = in_sizes[1] / 2;

    // workspace: h[N*D] | y[N*D] | dis[N]
    float* h   = (float*)d_ws;
    float* y   = h + (size_t)N * D;
    float* dis = y + (size_t)N * D;

    const int T   = 256;
    const int nND = N * D;
    dim3 gemmGrid((N + 16 * RT - 1) / (16 * RT), D / 16);

    // --- normalization ---
    k_init_deg <<<(N + T - 1) / T, T, 0, stream>>>(dis, N);
    k_count_deg<<<(E + T - 1) / T, T, 0, stream>>>(ei, dis, E);
    k_rsqrt    <<<(N + T - 1) / T, T, 0, stream>>>(dis, N);

    // --- layer 1 ---
    k_gemm_wmma_f32<<<gemmGrid, 32, 0, stream>>>(x, W1, h, N);
    k_init_out<<<(nND + T - 1) / T, T, 0, stream>>>(h, dis, b1, y, N);
    {
        long long nt = (long long)E * 32;
        k_scatter<<<(unsigned)((nt + T - 1) / T), T, 0, stream>>>(h, ei, dis, y, E);
    }
    k_relu<<<(nND + T - 1) / T, T, 0, stream>>>(y, nND);

    // --- layer 2 ---
    k_gemm_wmma_f32<<<gemmGrid, 32, 0, stream>>>(y, W2, h, N);
    k_init_out<<<(nND + T - 1) / T, T, 0, stream>>>(h, dis, b2, out, N);
    {
        long long nt = (long long)E * 32;
        k_scatter<<<(unsigned)((nt + T - 1) / T), T, 0, stream>>>(h, ei, dis, out, E);
    }
    k_relu<<<(nND + T - 1) / T, T, 0, stream>>>(out, nND);
}